// GNN_Node_6614249635914
// MI455X (gfx1250) — compile-verified
//
#include <hip/hip_runtime.h>
#include <hip/hip_bf16.h>

typedef float v2f __attribute__((ext_vector_type(2)));
typedef float v8f __attribute__((ext_vector_type(8)));

#define N_NODES_C 50000
#define N_EDGES_C 600000
#define EMBD 128
#define ZSTR 132          // LDS row stride (floats) to avoid bank conflicts
#define MB 64             // rows per GEMM block
#define BN_EPS_C 1e-5f

// ---------------- atom encoding: h[n,c] = sum_f atom_emb[f, x[n,f], c] ----------
__global__ __launch_bounds__(256) void atom_encode_kernel(
    const int* __restrict__ x, const float* __restrict__ atom_emb,
    float* __restrict__ h)
{
  int t = blockIdx.x * 256 + threadIdx.x;
  if (t >= N_NODES_C * EMBD) return;
  int n = t >> 7, c = t & 127;
  float acc = 0.f;
#pragma unroll
  for (int f = 0; f < 9; ++f) {
    int xi = x[n * 9 + f];
    acc += atom_emb[(f * 4 + xi) * EMBD + c];
  }
  h[t] = acc;
}

// ---------------- zero scratch ----------------
__global__ __launch_bounds__(256) void zero_kernel(float4* __restrict__ p, int n4)
{
  int t = blockIdx.x * 256 + threadIdx.x;
  if (t < n4) p[t] = make_float4(0.f, 0.f, 0.f, 0.f);
}

// ---------------- edge message + scatter add ----------------
// one wave (32 lanes) per edge; each lane handles 4 channels
__global__ __launch_bounds__(256) void edge_kernel(
    const float* __restrict__ h, const int* __restrict__ edge_index,
    const int* __restrict__ edge_attr, const float* __restrict__ bond_emb,
    float* __restrict__ agg, int layer)
{
  int t = blockIdx.x * 256 + threadIdx.x;
  int e = t >> 5;
  if (e >= N_EDGES_C) return;
  int c = (t & 31) << 2;
  int s  = edge_index[e];
  int d  = edge_index[N_EDGES_C + e];
  int a0 = edge_attr[e * 3 + 0];
  int a1 = edge_attr[e * 3 + 1];
  int a2 = edge_attr[e * 3 + 2];
  const float* B = bond_emb + (size_t)layer * 3 * 4 * EMBD;
  const float4 e0 = *(const float4*)&B[(0 * 4 + a0) * EMBD + c];
  const float4 e1 = *(const float4*)&B[(1 * 4 + a1) * EMBD + c];
  const float4 e2 = *(const float4*)&B[(2 * 4 + a2) * EMBD + c];
  const float4 hv = *(const float4*)&h[(size_t)s * EMBD + c];
  float m0 = fmaxf(hv.x + e0.x + e1.x + e2.x, 0.f);
  float m1 = fmaxf(hv.y + e0.y + e1.y + e2.y, 0.f);
  float m2 = fmaxf(hv.z + e0.z + e1.z + e2.z, 0.f);
  float m3 = fmaxf(hv.w + e0.w + e1.w + e2.w, 0.f);
  float* o = agg + (size_t)d * EMBD + c;
  atomicAdd(o + 0, m0);
  atomicAdd(o + 1, m1);
  atomicAdd(o + 2, m2);
  atomicAdd(o + 3, m3);
}

// ---------------- fused GIN MLP: z=(1+eps)h+agg; BN1(zW1+b1); relu; *W2+b2; BN2(relu); [relu] ----
// fp32 WMMA 16x16x4.  Per-lane fragment mapping (ISA 7.12.2):
//   A (16x4): row = lane%16, K = (lane/16)*2 + {0,1}   -> contiguous float2 from row-major LDS
//   B (4x16): col = lane%16, K = (lane/16)*2 + {0,1}   (mirror of A, N across lanes)
//   C/D (16x16): row = vgpr + 8*(lane/16), col = lane%16
__device__ __forceinline__ void gemm_16x128_k128(
    const float* __restrict__ ZT, const float* __restrict__ WL,
    int col, int arow, int khalf, v8f acc[4])
{
#pragma unroll 4
  for (int ks = 0; ks < 32; ++ks) {
    int k0 = (ks << 2) + khalf;
    v2f bf;
    bf.x = WL[k0 * ZSTR + col];
    bf.y = WL[(k0 + 1) * ZSTR + col];
#pragma unroll
    for (int mt = 0; mt < 4; ++mt) {
      v2f af = *(const v2f*)&ZT[(mt * 16 + arow) * ZSTR + k0];
      acc[mt] = __builtin_amdgcn_wmma_f32_16x16x4_f32(
          false, af, false, bf, (short)0, acc[mt], false, false);
    }
  }
}

__global__ __launch_bounds__(256) void gin_mlp_kernel(
    const float* __restrict__ h,  const float* __restrict__ agg,
    const float* __restrict__ eps,
    const float* __restrict__ W1, const float* __restrict__ b1,
    const float* __restrict__ g1, const float* __restrict__ be1,
    const float* __restrict__ m1, const float* __restrict__ v1,
    const float* __restrict__ W2, const float* __restrict__ b2,
    const float* __restrict__ g2, const float* __restrict__ be2,
    const float* __restrict__ m2, const float* __restrict__ v2,
    float* __restrict__ hout, int layer, int relu_out)
{
  extern __shared__ float smem[];
  float* ZT = smem;                 // MB x ZSTR   (Z tile, then T tile)
  float* WL = smem + MB * ZSTR;     // 128 x ZSTR  (W1, then W2)

  const int tid  = threadIdx.x;
  const int lane = tid & 31;
  const int wave = tid >> 5;                   // 0..7 -> N tile
  const int rowBase = blockIdx.x * MB;

  const int col   = wave * 16 + (lane & 15);   // 0..127, this lane's output column
  const int arow  = lane & 15;                 // A-fragment row within a 16-row tile
  const int khalf = (lane >> 4) << 1;          // 0 or 2

  const float ep = 1.0f + eps[layer];
  const float* W1l = W1 + (size_t)layer * EMBD * EMBD;
  const float* W2l = W2 + (size_t)layer * EMBD * EMBD;

  // ---- stage Z = (1+eps)*h + agg  (64x128) ----
#pragma unroll
  for (int i = 0; i < 8; ++i) {
    int idx = tid + i * 256;            // 0..2047
    int r = idx >> 5;
    int c4 = (idx & 31) << 2;
    int gr = rowBase + r;
    float4 z4 = make_float4(0.f, 0.f, 0.f, 0.f);
    if (gr < N_NODES_C) {
      const float4 hv = *(const float4*)&h[(size_t)gr * EMBD + c4];
      const float4 av = *(const float4*)&agg[(size_t)gr * EMBD + c4];
      z4.x = ep * hv.x + av.x;  z4.y = ep * hv.y + av.y;
      z4.z = ep * hv.z + av.z;  z4.w = ep * hv.w + av.w;
    }
    *(float4*)&ZT[r * ZSTR + c4] = z4;
  }
  // ---- stage W1 (128x128) ----
#pragma unroll
  for (int i = 0; i < 16; ++i) {
    int idx = tid + i * 256;            // 0..4095
    int r = idx >> 5;
    int c4 = (idx & 31) << 2;
    *(float4*)&WL[r * ZSTR + c4] = *(const float4*)&W1l[(size_t)r * EMBD + c4];
  }
  __syncthreads();

  // ---- GEMM1 ----
  v8f acc[4];
#pragma unroll
  for (int mt = 0; mt < 4; ++mt) acc[mt] = (v8f){0.f,0.f,0.f,0.f,0.f,0.f,0.f,0.f};
  gemm_16x128_k128(ZT, WL, col, arow, khalf, acc);
  __syncthreads();   // everyone done reading ZT (Z) and WL (W1)

  // ---- epilogue 1: +b1, BN1, relu  -> T into ZT ----
  {
    const float bia = b1[layer * EMBD + col];
    const float sc  = g1[layer * EMBD + col] * rsqrtf(v1[layer * EMBD + col] + BN_EPS_C);
    const float mu  = m1[layer * EMBD + col];
    const float bb  = be1[layer * EMBD + col];
    const int hoff = (lane >> 4) << 3;
#pragma unroll
    for (int mt = 0; mt < 4; ++mt)
#pragma unroll
      for (int r = 0; r < 8; ++r) {
        int m_loc = mt * 16 + hoff + r;
        float y = acc[mt][r] + bia;
        y = (y - mu) * sc + bb;
        ZT[m_loc * ZSTR + col] = fmaxf(y, 0.f);
      }
  }
  // ---- stage W2 ----
#pragma unroll
  for (int i = 0; i < 16; ++i) {
    int idx = tid + i * 256;
    int r = idx >> 5;
    int c4 = (idx & 31) << 2;
    *(float4*)&WL[r * ZSTR + c4] = *(const float4*)&W2l[(size_t)r * EMBD + c4];
  }
  __syncthreads();

  // ---- GEMM2 ----
#pragma unroll
  for (int mt = 0; mt < 4; ++mt) acc[mt] = (v8f){0.f,0.f,0.f,0.f,0.f,0.f,0.f,0.f};
  gemm_16x128_k128(ZT, WL, col, arow, khalf, acc);

  // ---- epilogue 2: +b2, relu, BN2, optional relu -> global ----
  {
    const float bia = b2[layer * EMBD + col];
    const float sc  = g2[layer * EMBD + col] * rsqrtf(v2[layer * EMBD + col] + BN_EPS_C);
    const float mu  = m2[layer * EMBD + col];
    const float bb  = be2[layer * EMBD + col];
    const int hoff = (lane >> 4) << 3;
#pragma unroll
    for (int mt = 0; mt < 4; ++mt)
#pragma unroll
      for (int r = 0; r < 8; ++r) {
        int m_loc = mt * 16 + hoff + r;
        int gr = rowBase + m_loc;
        float y = acc[mt][r] + bia;
        y = fmaxf(y, 0.f);              // relu before BN2
        y = (y - mu) * sc + bb;         // BN2
        if (relu_out) y = fmaxf(y, 0.f);
        if (gr < N_NODES_C) hout[(size_t)gr * EMBD + col] = y;
      }
  }
}

extern "C" void kernel_launch(void* const* d_in, const int* in_sizes, int n_in,
                              void* d_out, int out_size, void* d_ws, size_t ws_size,
                              hipStream_t stream) {
  const int*   x         = (const int*)  d_in[0];
  const int*   edge_idx  = (const int*)  d_in[1];
  const int*   edge_attr = (const int*)  d_in[2];
  const float* atom_emb  = (const float*)d_in[3];
  const float* bond_emb  = (const float*)d_in[4];
  const float* eps       = (const float*)d_in[5];
  const float* W1        = (const float*)d_in[6];
  const float* b1        = (const float*)d_in[7];
  const float* g1        = (const float*)d_in[8];
  const float* be1       = (const float*)d_in[9];
  const float* m1        = (const float*)d_in[10];
  const float* v1        = (const float*)d_in[11];
  const float* W2        = (const float*)d_in[12];
  const float* b2        = (const float*)d_in[13];
  const float* g2        = (const float*)d_in[14];
  const float* be2       = (const float*)d_in[15];
  const float* m2        = (const float*)d_in[16];
  const float* v2        = (const float*)d_in[17];
  float* out = (float*)d_out;

  float* agg = (float*)d_ws;                              // 50000*128 floats
  float* hB  = agg + (size_t)N_NODES_C * EMBD;            // 50000*128 floats

  const int hElems = N_NODES_C * EMBD;                    // 6,400,000
  const int zeroBlocks = (hElems / 4 + 255) / 256;        // float4 zeroing
  const int atomBlocks = (hElems + 255) / 256;
  const int edgeBlocks = (N_EDGES_C * 32 + 255) / 256;
  const int gemmBlocks = (N_NODES_C + MB - 1) / MB;
  const size_t smemBytes = (size_t)(MB * ZSTR + EMBD * ZSTR) * sizeof(float); // 101376

  // atom encoding -> hB
  atom_encode_kernel<<<atomBlocks, 256, 0, stream>>>(x, atom_emb, hB);

  for (int l = 0; l < 5; ++l) {
    const float* hcur = (l % 2 == 0) ? hB : out;
    float*       hnxt = (l % 2 == 0) ? out : hB;
    // zero agg
    zero_kernel<<<zeroBlocks, 256, 0, stream>>>((float4*)agg, hElems / 4);
    // messages + scatter
    edge_kernel<<<edgeBlocks, 256, 0, stream>>>(hcur, edge_idx, edge_attr,
                                                bond_emb, agg, l);
    // fused GIN MLP
    gin_mlp_kernel<<<gemmBlocks, 256, smemBytes, stream>>>(
        hcur, agg, eps, W1, b1, g1, be1, m1, v1, W2, b2, g2, be2, m2, v2,
        hnxt, l, (l < 4) ? 1 : 0);
  }
  // layer 4 writes `out` (hB -> out), so d_out holds the final h
}